// MessagePassingNodeModule_44942537785403
// MI455X (gfx1250) — compile-verified
//
#include <hip/hip_runtime.h>
#include <hip/hip_bf16.h>

#define N_NODES   100000
#define N_EDGES   1600000
#define D         128
#define IN_FEAT   384

typedef __attribute__((ext_vector_type(2))) float v2f;
typedef __attribute__((ext_vector_type(4))) float v4f;
typedef __attribute__((ext_vector_type(8))) float v8f;

// ---------------- workspace layout (floats) ----------------
// [0, N_NODES*D)                : summed edge features
// [N_NODES*D, N_NODES*D+N_NODES): per-node edge counts (as float)
#define WS_SUMMED_OFF 0
#define WS_COUNT_OFF  (N_NODES * D)
#define WS_TOTAL      (N_NODES * D + N_NODES)   // 12,900,000 floats (mult of 4)

// ---------------- kernel 1: zero the workspace ----------------
__global__ void zero_ws_kernel(float* __restrict__ ws, int n4) {
    int i = blockIdx.x * blockDim.x + threadIdx.x;
    if (i < n4) {
        v4f z = {0.f, 0.f, 0.f, 0.f};
        ((v4f*)ws)[i] = z;
    }
}

// ---------------- kernel 2: scatter-add edges ----------------
// One wave32 per edge. Each lane handles 4 consecutive channels (128-bit NT load).
// Non-temporal load of edge_attr keeps the 51MB accumulator resident in L2.
__global__ void edge_scatter_kernel(const float* __restrict__ edge_attr,
                                    const int* __restrict__ dest,
                                    float* __restrict__ summed,
                                    float* __restrict__ counts) {
    int gid  = blockIdx.x * blockDim.x + threadIdx.x;
    int edge = gid >> 5;
    int lane = threadIdx.x & 31;
    if (edge >= N_EDGES) return;

    int d = dest[edge];
    const v4f* src = (const v4f*)(edge_attr + (size_t)edge * D) + lane;
    v4f v = __builtin_nontemporal_load(src);

    float* dst = summed + (size_t)d * D + lane * 4;
    unsafeAtomicAdd(dst + 0, v.x);
    unsafeAtomicAdd(dst + 1, v.y);
    unsafeAtomicAdd(dst + 2, v.z);
    unsafeAtomicAdd(dst + 3, v.w);
    if (lane == 0) unsafeAtomicAdd(counts + d, 1.0f);
}

// ---------------- kernel 3: fused node MLP via WMMA f32 ----------------
// Block = 256 threads = 8 waves, handles a tile of 16 nodes.
// Wave w owns output columns [16w, 16w+16).
// GEMM1: feats[16x384] @ W1[384x128]  (96 chained v_wmma_f32_16x16x4_f32 per wave)
// GEMM2: relu(h)[16x128] @ W2[128x128] (32 chained WMMAs per wave)
#define FS 388   // LDS row stride for feats (388 % 64 == 4 -> conflict-free A reads)
#define HS 132   // LDS row stride for hidden (132 % 64 == 4)

__global__ __launch_bounds__(256)
void node_mlp_wmma_kernel(const float* __restrict__ x,
                          const float* __restrict__ u,
                          const float* __restrict__ W1,
                          const float* __restrict__ b1,
                          const float* __restrict__ W2,
                          const float* __restrict__ b2,
                          const float* __restrict__ summed,
                          const float* __restrict__ counts,
                          const int*   __restrict__ batch,
                          float*       __restrict__ out) {
    __shared__ float feats[16 * FS];   // 24832 B
    __shared__ float hbuf [16 * HS];   //  8448 B

    const int tile = blockIdx.x;
    const int base = tile * 16;
    const int tid  = threadIdx.x;
    const int lane = tid & 31;
    const int wave = tid >> 5;

    // ---- stage feats tile: [x | summed/max(cnt,1) | u[batch]] ----
    #pragma unroll
    for (int i = 0; i < 2; ++i) {
        int row  = wave * 2 + i;
        int node = base + row;
        float inv = 1.0f / fmaxf(counts[node], 1.0f);
        int g = batch[node];
        for (int c = lane; c < D; c += 32) {
            feats[row * FS + c]           = x[(size_t)node * D + c];
            feats[row * FS + 128 + c]     = summed[(size_t)node * D + c] * inv;
            feats[row * FS + 256 + c]     = u[g * D + c];
        }
    }
    __syncthreads();

    const int n    = wave * 16 + (lane & 15);   // output column this lane owns
    const int arow = lane & 15;                 // A-matrix row (node within tile)
    const int koff = (lane >> 4) * 2;           // K sub-offset per ISA A layout

    // ---- GEMM1: 16x384 @ 384x128 tile, accumulate over K in steps of 4 ----
    v8f c1 = {0.f, 0.f, 0.f, 0.f, 0.f, 0.f, 0.f, 0.f};
    for (int k = 0; k < IN_FEAT; k += 4) {
        v2f a = *(const v2f*)&feats[arow * FS + k + koff];
        v2f b;
        b.x = W1[(k + koff)     * D + n];
        b.y = W1[(k + koff + 1) * D + n];
        c1 = __builtin_amdgcn_wmma_f32_16x16x4_f32(false, a, false, b,
                                                   (short)0, c1, false, false);
    }

    // ---- bias + ReLU, write hidden tile to LDS ----
    float bias1 = b1[n];
    #pragma unroll
    for (int r = 0; r < 8; ++r) {
        int m = r + (lane >> 4) * 8;           // C/D layout: lanes 16-31 hold M+8
        hbuf[m * HS + n] = fmaxf(c1[r] + bias1, 0.0f);
    }
    __syncthreads();

    // ---- GEMM2: 16x128 @ 128x128 tile ----
    v8f c2 = {0.f, 0.f, 0.f, 0.f, 0.f, 0.f, 0.f, 0.f};
    for (int k = 0; k < D; k += 4) {
        v2f a = *(const v2f*)&hbuf[arow * HS + k + koff];
        v2f b;
        b.x = W2[(k + koff)     * D + n];
        b.y = W2[(k + koff + 1) * D + n];
        c2 = __builtin_amdgcn_wmma_f32_16x16x4_f32(false, a, false, b,
                                                   (short)0, c2, false, false);
    }

    // ---- bias + store ----
    float bias2 = b2[n];
    #pragma unroll
    for (int r = 0; r < 8; ++r) {
        int m = r + (lane >> 4) * 8;
        out[(size_t)(base + m) * D + n] = c2[r] + bias2;
    }
}

extern "C" void kernel_launch(void* const* d_in, const int* in_sizes, int n_in,
                              void* d_out, int out_size, void* d_ws, size_t ws_size,
                              hipStream_t stream) {
    const float* x         = (const float*)d_in[0];
    const float* edge_attr = (const float*)d_in[1];
    const float* u         = (const float*)d_in[2];
    const float* W1        = (const float*)d_in[3];
    const float* b1        = (const float*)d_in[4];
    const float* W2        = (const float*)d_in[5];
    const float* b2        = (const float*)d_in[6];
    const int*   edge_idx  = (const int*)d_in[7];   // [2, N_EDGES]; row 1 = dest
    const int*   batch     = (const int*)d_in[8];
    float*       out       = (float*)d_out;

    float* ws      = (float*)d_ws;
    float* summed  = ws + WS_SUMMED_OFF;
    float* counts  = ws + WS_COUNT_OFF;
    const int* dest = edge_idx + N_EDGES;

    // 1) zero accumulators
    {
        int n4 = WS_TOTAL / 4;
        int blocks = (n4 + 255) / 256;
        zero_ws_kernel<<<blocks, 256, 0, stream>>>(ws, n4);
    }
    // 2) scatter-add edge features (one wave per edge)
    {
        long long threads = (long long)N_EDGES * 32;
        int blocks = (int)((threads + 255) / 256);
        edge_scatter_kernel<<<blocks, 256, 0, stream>>>(edge_attr, dest, summed, counts);
    }
    // 3) fused feature-build + MLP (WMMA)
    {
        int blocks = N_NODES / 16;   // 6250, exact
        node_mlp_wmma_kernel<<<blocks, 256, 0, stream>>>(x, u, W1, b1, W2, b2,
                                                         summed, counts, batch, out);
    }
}